// HDCNN_15367392985642
// MI455X (gfx1250) — compile-verified
//
#include <hip/hip_runtime.h>

// HDCNN hyperbolic expansive conv stack for gfx1250 (MI455X).
// Layer = banded-GEMM conv via v_wmma_f32_16x16x4_f32, per-row reduction
// (collapses expmap0/proj/mobius_add to two scalars), fused pointwise ReLU.

typedef float v2f __attribute__((ext_vector_type(2)));
typedef float v8f __attribute__((ext_vector_type(8)));

#define HD_BATCH 32768
#define HD_FLEN  64
#define HD_MAXL  1280

// ---------------------------------------------------------------------------
// Conv kernel: each wave32 computes 4 consecutive 16(batch) x 16(position)
// output tiles (one batch tile x 64 output columns).
// Full conv y[b, j0+n] = sum_{kt=0..79} v[b, j0-64+kt] * w[n+64-kt]
// => per tile: M=16, N=16, K=80 -> 20 chained WMMA_F32_16X16X4_F32.
// A (16x4 f32): lanes 0-15 hold M=0..15 @ K={0,1}; lanes 16-31 @ K={2,3}.
// B (4x16 f32): lanes 0-15 hold N=0..15 @ K={0,1}; lanes 16-31 @ K={2,3}.
// D (16x16 f32): reg r, lane l -> M = r + 8*(l>>4), N = l&15.
//
// Branch-free data movement:
//  * Filter staged zero-padded into LDS (wp[96], taps at [16,80)) so every
//    B-tile read is an unguarded ds_load -- no compares, no EXEC branches.
//  * Interior tiles: unguarded global b64 loads (t0 even, rows 64B-aligned).
//  * Boundary tiles: clamped-address unconditional b64 load + v_cndmask
//    zeroing (one compare covers both components since t0 and Lin are even).
// All branches that remain are wave-uniform, so EXEC is all-ones at WMMA.
// ---------------------------------------------------------------------------
__global__ __launch_bounds__(256) void hdcnn_conv_wmma(
    const float* __restrict__ h, const float* __restrict__ w,
    float* __restrict__ conv, int Lin, int Lout, int numJG)
{
  __shared__ float wp[96];                   // zero-padded filter
  const int tid = threadIdx.x;
  if (tid < 96) {
    wp[tid] = (tid >= 16 && tid < 16 + HD_FLEN) ? w[tid - 16] : 0.0f;
  }
  __syncthreads();

  const int lane  = tid & 31;
  const int wib   = tid >> 5;
  const int jg    = blockIdx.x * 8 + wib;    // group of 4 position tiles
  if (jg >= numJG) return;                   // wave-uniform exit (before WMMA)
  const int bt    = blockIdx.y;              // batch tile
  const int m     = lane & 15;               // batch row (A) / position n (B,D)
  const int khalf = lane >> 4;               // K half select

  const float* __restrict__ hrow = h + (size_t)(bt * 16 + m) * Lin;

  // Filter tiles B_g (position-tile invariant): unguarded LDS reads.
  v2f b[20];
#pragma unroll
  for (int g = 0; g < 20; ++g) {
    const int kt0 = 4 * g + 2 * khalf;
    const int idx = m + 80 - kt0;            // in [2, 95]; idx-1 in [1, 94]
    b[g].x = wp[idx];
    b[g].y = wp[idx - 1];
  }

  for (int s = 0; s < 4; ++s) {
    const int j0    = (jg * 4 + s) << 4;
    const int tbase = j0 - 64 + 2 * khalf;   // even
    v8f acc = {};

    if (j0 >= 64 && j0 + 16 <= Lin) {
      // Interior: all taps in range -> unguarded 8-byte loads.
      const v2f* __restrict__ hv = (const v2f*)(hrow + tbase);
#pragma unroll
      for (int g = 0; g < 20; ++g) {
        acc = __builtin_amdgcn_wmma_f32_16x16x4_f32(false, hv[2 * g], false,
                                                    b[g], (short)0, acc,
                                                    false, false);
      }
    } else {
      // Boundary: clamp address, unconditional load, zero via selects.
#pragma unroll
      for (int g = 0; g < 20; ++g) {
        const int  t0  = tbase + 4 * g;      // even
        const bool ok  = (unsigned)t0 < (unsigned)Lin;  // covers t0 and t0+1
        const int  t0c = ok ? t0 : 0;
        const v2f  av  = *(const v2f*)(hrow + t0c);
        v2f a;
        a.x = ok ? av.x : 0.0f;
        a.y = ok ? av.y : 0.0f;
        acc = __builtin_amdgcn_wmma_f32_16x16x4_f32(false, a, false, b[g],
                                                    (short)0, acc,
                                                    false, false);
      }
    }

    // Store D: reg r -> batch row bt*16 + r + 8*khalf, column j0 + (lane&15).
    float* __restrict__ crow0 =
        conv + (size_t)(bt * 16 + 8 * khalf) * Lout + j0 + m;
#pragma unroll
    for (int r = 0; r < 8; ++r) {
      crow0[(size_t)r * Lout] = acc[r];
    }
  }
}

// ---------------------------------------------------------------------------
// Per-row reduction. Computes sumsq(conv), <conv,bias>, sumsq(bias), then
// folds expmap0 + proj + mobius_add coefficients into two scalars p,q:
//   out[j] = relu(p*conv[j] + q*bias[j])
// ---------------------------------------------------------------------------
__global__ __launch_bounds__(256) void hdcnn_row_reduce(
    const float* __restrict__ conv, const float* __restrict__ bias,
    float* __restrict__ pq, int Lout)
{
  __shared__ float s0[256], s1[256], s2[256];
  const int tid = threadIdx.x;
  const float* __restrict__ c = conv + (size_t)blockIdx.x * Lout;

  float ss = 0.0f, dd = 0.0f, bb = 0.0f;
  for (int j = tid; j < Lout; j += 256) {
    const float cv = c[j];
    const float bv = bias[j];
    ss = fmaf(cv, cv, ss);
    dd = fmaf(cv, bv, dd);
    bb = fmaf(bv, bv, bb);
  }
  s0[tid] = ss; s1[tid] = dd; s2[tid] = bb;
  __syncthreads();
#pragma unroll
  for (int off = 128; off > 0; off >>= 1) {
    if (tid < off) {
      s0[tid] += s0[tid + off];
      s1[tid] += s1[tid + off];
      s2[tid] += s2[tid + off];
    }
    __syncthreads();
  }

  if (tid == 0) {
    const float sumsq = s0[0], dot = s1[0], y2 = s2[0];
    const float n = fmaxf(sqrtf(sumsq), 1e-15f);        // _norm clamp
    const float t = tanhf(n);                           // expmap0 (c=1)
    const float xnorm = fmaxf(t, 1e-15f);               // ||x|| after expmap0
    const float maxnorm = 1.0f - 4e-3f;                 // proj boundary
    const float scale = (xnorm > maxnorm) ? (maxnorm / xnorm) : 1.0f;
    const float alpha = (t / n) * scale;                // x = alpha * conv
    const float x2 = alpha * alpha * sumsq;
    const float xy = alpha * dot;
    const float cA = 1.0f + 2.0f * xy + y2;             // (1+2c*xy+c*y2)
    const float cB = 1.0f - x2;                         // (1-c*x2)
    const float den = fmaxf(1.0f + 2.0f * xy + x2 * y2, 1e-15f);
    pq[2 * blockIdx.x]     = cA * alpha / den;          // p
    pq[2 * blockIdx.x + 1] = cB / den;                  // q
  }
}

// ---------------------------------------------------------------------------
// Fused mobius_add + ReLU (pointwise, per-row scalars). Safe to run in-place
// on the conv buffer (same-index read-then-write).
// ---------------------------------------------------------------------------
__global__ __launch_bounds__(256) void hdcnn_pointwise(
    const float* __restrict__ conv, const float* __restrict__ bias,
    const float* __restrict__ pq, float* __restrict__ out, int Lout)
{
  const int j   = blockIdx.x * 256 + threadIdx.x;
  const int row = blockIdx.y;
  if (j >= Lout) return;
  const size_t idx = (size_t)row * Lout + j;
  const float v = pq[2 * row] * conv[idx] + pq[2 * row + 1] * bias[j];
  out[idx] = fmaxf(v, 0.0f);
}

// ---------------------------------------------------------------------------
extern "C" void kernel_launch(void* const* d_in, const int* in_sizes, int n_in,
                              void* d_out, int out_size, void* d_ws, size_t ws_size,
                              hipStream_t stream)
{
  (void)in_sizes; (void)n_in; (void)out_size; (void)ws_size;

  const float* hk = (const float*)d_in[0];
  const float* w  = (const float*)d_in[1];
  const float* bias[4] = {(const float*)d_in[2], (const float*)d_in[3],
                          (const float*)d_in[4], (const float*)d_in[5]};

  // Workspace: two ping-pong [32768, 1280] fp32 buffers + per-row (p,q).
  float* bufA = (float*)d_ws;
  float* bufB = bufA + (size_t)HD_BATCH * HD_MAXL;
  float* pq   = bufB + (size_t)HD_BATCH * HD_MAXL;

  const float* hin = hk;
  int Lin = 1024;
  for (int i = 0; i < 4; ++i) {
    const int Lout = Lin + HD_FLEN;                 // multiple of 64
    float* conv = (i & 1) ? bufB : bufA;            // ping-pong; never aliases hin
    const int numJG = Lout / 64;                    // groups of 4 position tiles

    dim3 gc((numJG + 7) / 8, HD_BATCH / 16);
    hdcnn_conv_wmma<<<gc, 256, 0, stream>>>(hin, w + i * HD_FLEN,
                                            conv, Lin, Lout, numJG);
    hdcnn_row_reduce<<<HD_BATCH, 256, 0, stream>>>(conv, bias[i], pq, Lout);

    float* dst = (i == 3) ? (float*)d_out : conv;   // in-place elementwise OK
    dim3 gp((Lout + 255) / 256, HD_BATCH);
    hdcnn_pointwise<<<gp, 256, 0, stream>>>(conv, bias[i], pq, dst, Lout);

    hin = dst;
    Lin = Lout;
  }
}